// DeepClusterLoss_7507602833888
// MI455X (gfx1250) — compile-verified
//
#include <hip/hip_runtime.h>

// DeepClusterLoss for MI455X (gfx1250, wave32).
// Memory-bound streaming reduction: ~520 MB @ 23.3 TB/s => ~22 us floor.
// - b128 (float4) global loads for recon_x / x
// - centers staged in LDS (25.6 KB of 320 KB/WGP), gathered via ds_load_b128
// - intra-wave reduction done with V_WMMA_F32_16X16X4_F32 (ones-matrix trick)
// - global_prefetch_b8 for the next grid-stride chunk
// - deterministic two-pass reduction (no float atomics)

#define ALPHA 1.0f
#define BETA  1.0f
#define D_DIM 64
#define D4    16            // D_DIM / 4 (float4 granularity)
#define MAX_K 128           // K = 100 in the reference; clamp guard below
#define NB    2048          // fixed block count -> every d_ws slot written
#define NT    256           // 8 waves of 32

typedef __attribute__((ext_vector_type(2))) float v2f;
typedef __attribute__((ext_vector_type(8))) float v8f;

// Full 32-lane sum using one V_WMMA_F32_16X16X4_F32.
// A (16x4 f32, 2 VGPRs/lane): VGPR0 = lane value (K=0 for lanes 0-15, K=2 for
// lanes 16-31), VGPR1 = 0. B = all-ones 4x16. Then for any column n:
//   D[m][n] = A[m][0] + A[m][2] = v_m + v_{m+16}
// Lane l<16 holds D[0..7][l] in the 8 C VGPRs, lane l+16 holds D[8..15][l].
// Sum of the 8 VGPRs + shfl_xor(16) = sum over all 32 lanes, in every lane.
__device__ __forceinline__ float wave_sum_wmma(float v) {
    v2f a; a[0] = v;    a[1] = 0.0f;
    v2f b; b[0] = 1.0f; b[1] = 1.0f;
    v8f c = {};
    c = __builtin_amdgcn_wmma_f32_16x16x4_f32(
            /*neg_a=*/false, a, /*neg_b=*/false, b,
            /*c_mod=*/(short)0, c, /*reuse_a=*/false, /*reuse_b=*/false);
    float s = c[0] + c[1] + c[2] + c[3] + c[4] + c[5] + c[6] + c[7];
    s += __shfl_xor(s, 16, 32);
    return s;
}

__global__ void __launch_bounds__(NT)
dcl_partial_kernel(const float* __restrict__ recon,
                   const float* __restrict__ x,
                   const int*   __restrict__ assign_i64_lo, // int64 viewed as int pairs
                   const float* __restrict__ centers,
                   float* __restrict__ partial,             // [NB][2]
                   int n4, int K) {
    __shared__ float s_cent[MAX_K * D_DIM];                 // 32 KB max
    const int kd = (K > MAX_K ? MAX_K : K) * D_DIM;
    for (int i = threadIdx.x; i < kd; i += NT) s_cent[i] = centers[i];
    __syncthreads();

    const float4* r4 = (const float4*)recon;
    const float4* x4 = (const float4*)x;
    const float4* c4 = (const float4*)s_cent;

    float accR = 0.0f, accC = 0.0f;
    const int stride = gridDim.x * NT;
    for (int i = blockIdx.x * NT + threadIdx.x; i < n4; i += stride) {
        // prefetch next grid-stride chunk (speculative; OOB is silently dropped)
        __builtin_prefetch(&r4[i + stride], 0, 1);
        __builtin_prefetch(&x4[i + stride], 0, 1);

        float4 rv = r4[i];
        float4 xv = x4[i];
        const int row = i >> 4;            // i / D4
        const int col = i & (D4 - 1);
        const int cid = assign_i64_lo[2 * row];   // low dword of int64, value < K
        float4 cv = c4[cid * D4 + col];           // ds_load_b128 gather from LDS

        float d0 = rv.x - xv.x, d1 = rv.y - xv.y, d2 = rv.z - xv.z, d3 = rv.w - xv.w;
        accR = fmaf(d0, d0, accR); accR = fmaf(d1, d1, accR);
        accR = fmaf(d2, d2, accR); accR = fmaf(d3, d3, accR);

        float e0 = xv.x - cv.x, e1 = xv.y - cv.y, e2 = xv.z - cv.z, e3 = xv.w - cv.w;
        accC = fmaf(e0, e0, accC); accC = fmaf(e1, e1, accC);
        accC = fmaf(e2, e2, accC); accC = fmaf(e3, e3, accC);
    }

    // Wave-level reduction on the matrix pipe (all 32 lanes active here).
    float wr = wave_sum_wmma(accR);
    float wc = wave_sum_wmma(accC);

    __shared__ float s_red[2][NT / 32];
    const int wave = threadIdx.x >> 5;
    const int lane = threadIdx.x & 31;
    if (lane == 0) { s_red[0][wave] = wr; s_red[1][wave] = wc; }
    __syncthreads();
    if (threadIdx.x == 0) {
        float tr = 0.0f, tc = 0.0f;
        #pragma unroll
        for (int w = 0; w < NT / 32; ++w) { tr += s_red[0][w]; tc += s_red[1][w]; }
        partial[2 * blockIdx.x + 0] = tr;
        partial[2 * blockIdx.x + 1] = tc;
    }
}

__global__ void __launch_bounds__(NT)
dcl_final_kernel(const float* __restrict__ partial, float* __restrict__ out, int nb) {
    float r = 0.0f, c = 0.0f;
    for (int i = threadIdx.x; i < nb; i += NT) {
        r += partial[2 * i + 0];
        c += partial[2 * i + 1];
    }
    r = wave_sum_wmma(r);
    c = wave_sum_wmma(c);

    __shared__ float sr[NT / 32], sc[NT / 32];
    const int wave = threadIdx.x >> 5;
    const int lane = threadIdx.x & 31;
    if (lane == 0) { sr[wave] = r; sc[wave] = c; }
    __syncthreads();
    if (threadIdx.x == 0) {
        float tr = 0.0f, tc = 0.0f;
        #pragma unroll
        for (int w = 0; w < NT / 32; ++w) { tr += sr[w]; tc += sc[w]; }
        out[0] = ALPHA * tr + BETA * tc;   // total_loss
        out[1] = tr;                       // recon_loss
        out[2] = tc;                       // cluster_loss
    }
}

extern "C" void kernel_launch(void* const* d_in, const int* in_sizes, int n_in,
                              void* d_out, int out_size, void* d_ws, size_t ws_size,
                              hipStream_t stream) {
    const float* recon   = (const float*)d_in[0];   // [N, D] fp32
    const float* x       = (const float*)d_in[1];   // [N, D] fp32
    const int*   assign  = (const int*)  d_in[2];   // [N] int64 -> int pairs (LE)
    const float* centers = (const float*)d_in[3];   // [K, D] fp32
    float* out = (float*)d_out;                     // [3] fp32

    const int nd = in_sizes[0];                     // N * D
    const int n4 = nd / 4;                          // float4 count
    const int K  = in_sizes[3] / D_DIM;

    float* partial = (float*)d_ws;                  // NB * 2 floats = 16 KB

    dcl_partial_kernel<<<NB, NT, 0, stream>>>(recon, x, assign, centers,
                                              partial, n4, K);
    dcl_final_kernel<<<1, NT, 0, stream>>>(partial, out, NB);
}